// Covid19CasesPredictModel_36155034698446
// MI455X (gfx1250) — compile-verified
//
#include <hip/hip_runtime.h>
#include <math.h>

// Covid19 cases model: b[t] = relu(a.x[t]); linear 20-tap recurrence
//   n[t]  = b[t] * sum_j surv[j]*n[t-20+j]
//   out[t]=        sum_j dsurv[j]*n[t-20+j]
// Parallelized as an exact chunked linear scan (recurrence is linear in the
// state window): per-chunk 20x20 transfer matrices -> WMMA f32 prefix scan
// -> per-chunk replay from exact boundary states.

#define T_STEPS 8192
#define LATENCY 21
#define W       20           // window length
#define CHUNK   64           // steps per chunk
#define NCHUNK  (T_STEPS / CHUNK)   // 128

typedef __attribute__((ext_vector_type(2))) float v2f;
typedef __attribute__((ext_vector_type(8))) float v8f;

// surv (already reversed, length 21), matching the reference _survival_rate.
__device__ __forceinline__ void make_surv(float h, float* surv) {
    float e = expf(-h);
    float s[LATENCY];
    s[0] = 1.0f;
#pragma unroll
    for (int t = 1; t < LATENCY; ++t) s[t] = s[t - 1] * e;
    float last = s[LATENCY - 1];
    float inv  = 1.0f / (1.0f - last);
#pragma unroll
    for (int j = 0; j < LATENCY; ++j) surv[j] = (s[LATENCY - 1 - j] - last) * inv;
}

// ---------------- K1: forcing term b[t] = relu(a . x[t,:]) ----------------
__global__ void k_forcing(const float* __restrict__ x, const float* __restrict__ a,
                          float* __restrict__ b) {
    int t = blockIdx.x * blockDim.x + threadIdx.x;
    if (t >= T_STEPS) return;
    float acc = 0.0f;
#pragma unroll
    for (int p = 0; p < 17; ++p) acc = fmaf(a[p], x[t * 17 + p], acc);
    b[t] = fmaxf(acc, 0.0f);
}

// ---------------- K2: per-chunk 20x20 transfer matrix (padded 32x32) -------
// One wave per chunk. Lane j runs the recurrence from basis window e_j; the
// final window is column j of G_c. Lanes j>=20 naturally produce zeros, and
// every lane writes its full 32-row column, so G comes out fully initialized.
__global__ void k_chunk_transfer(const float* __restrict__ b, const float* __restrict__ hp,
                                 float* __restrict__ G) {
    int c = blockIdx.x;      // chunk id
    int j = threadIdx.x;     // column 0..31
    float surv[LATENCY];
    make_surv(*hp, surv);
    float win[W];
#pragma unroll
    for (int i = 0; i < W; ++i) win[i] = (i == j) ? 1.0f : 0.0f;
    const float* bc = b + c * CHUNK;
    for (int t = 0; t < CHUNK; ++t) {
        float dot = 0.0f;
#pragma unroll
        for (int i = 0; i < W; ++i) dot = fmaf(surv[i], win[i], dot);
        float n = bc[t] * dot;
#pragma unroll
        for (int i = 0; i < W - 1; ++i) win[i] = win[i + 1];
        win[W - 1] = n;
    }
    float* Gc = G + c * 1024;        // 32x32 row-major
#pragma unroll
    for (int r = 0; r < W; ++r)  Gc[r * 32 + j] = win[r];
#pragma unroll
    for (int r = W; r < 32; ++r) Gc[r * 32 + j] = 0.0f;
}

// ---------------- K3: WMMA prefix scan of chunk matrices -------------------
// P <- I; for each chunk: S_c = P[:,19] (init window is e_19), P <- G_c * P.
// 32x32 product = 2x2 tiles of v_wmma_f32_16x16x4_f32, K=32 as 8 chained WMMAs.
__global__ __launch_bounds__(128) void k_scan(const float* __restrict__ G,
                                              float* __restrict__ S) {
    __shared__ float Pm[1024];   // prefix product (32x32)
    __shared__ float Gm[1024];   // current chunk matrix
    __shared__ float Nm[1024];   // product staging
    int tid = threadIdx.x;
#pragma unroll
    for (int k = 0; k < 8; ++k) {
        int idx = tid + 128 * k;
        Pm[idx] = ((idx >> 5) == (idx & 31)) ? 1.0f : 0.0f;   // identity
    }
    __syncthreads();

    int lane = tid & 31;
    int wave = tid >> 5;
    int ti   = (wave >> 1) * 16;   // output tile row base
    int tj   = (wave & 1) * 16;    // output tile col base
    int half = lane >> 4;          // 0: K pair {0,1}, 1: K pair {2,3}
    int l15  = lane & 15;

    for (int c = 0; c < NCHUNK; ++c) {
        if (tid < W) S[c * 32 + tid] = Pm[tid * 32 + (W - 1)];   // boundary state
#pragma unroll
        for (int k = 0; k < 8; ++k) Gm[tid + 128 * k] = G[c * 1024 + tid + 128 * k];
        __syncthreads();

        v8f acc = {};
#pragma unroll
        for (int kb = 0; kb < 8; ++kb) {
            int kk = kb * 4 + half * 2;
            v2f av, bv;
            av.x = Gm[(ti + l15) * 32 + kk];        // A = G_c, 16x4 fragment
            av.y = Gm[(ti + l15) * 32 + kk + 1];
            bv.x = Pm[kk * 32 + tj + l15];          // B = P, 4x16 fragment
            bv.y = Pm[(kk + 1) * 32 + tj + l15];
            acc = __builtin_amdgcn_wmma_f32_16x16x4_f32(
                false, av, false, bv, (short)0, acc, false, false);
        }
#pragma unroll
        for (int r = 0; r < 8; ++r)
            Nm[(ti + r + 8 * half) * 32 + tj + l15] = acc[r];
        __syncthreads();
#pragma unroll
        for (int k = 0; k < 8; ++k) Pm[tid + 128 * k] = Nm[tid + 128 * k];
        __syncthreads();
    }
}

// ---------------- K4: replay each chunk from its exact boundary state ------
__global__ void k_emit(const float* __restrict__ b, const float* __restrict__ S,
                       const float* __restrict__ hp, float* __restrict__ out) {
    int c = blockIdx.x * blockDim.x + threadIdx.x;
    if (c >= NCHUNK) return;
    float surv[LATENCY];
    make_surv(*hp, surv);
    float dsurv[W];
#pragma unroll
    for (int j = 0; j < W; ++j) dsurv[j] = surv[j + 1] - surv[j];
    float win[W];
#pragma unroll
    for (int i = 0; i < W; ++i) win[i] = S[c * 32 + i];
    const float* bc = b + c * CHUNK;
    float*       oc = out + c * CHUNK;
    for (int t = 0; t < CHUNK; ++t) {
        float o = 0.0f, dot = 0.0f;
#pragma unroll
        for (int i = 0; i < W; ++i) {
            o   = fmaf(dsurv[i], win[i], o);
            dot = fmaf(surv[i],  win[i], dot);
        }
        oc[t] = o;
        float n = bc[t] * dot;
#pragma unroll
        for (int i = 0; i < W - 1; ++i) win[i] = win[i + 1];
        win[W - 1] = n;
    }
}

extern "C" void kernel_launch(void* const* d_in, const int* in_sizes, int n_in,
                              void* d_out, int out_size, void* d_ws, size_t ws_size,
                              hipStream_t stream) {
    (void)in_sizes; (void)n_in; (void)out_size; (void)ws_size;
    const float* x = (const float*)d_in[0];   // (8192, 17)
    const float* a = (const float*)d_in[1];   // (17,)
    const float* h = (const float*)d_in[2];   // scalar
    float* out = (float*)d_out;               // (8192,)

    float* ws = (float*)d_ws;
    float* b  = ws;                           // 8192 floats
    float* G  = ws + T_STEPS;                 // NCHUNK * 1024 floats (32x32 each)
    float* S  = G + NCHUNK * 1024;            // NCHUNK * 32 floats

    k_forcing<<<T_STEPS / 256, 256, 0, stream>>>(x, a, b);
    k_chunk_transfer<<<NCHUNK, 32, 0, stream>>>(b, h, G);
    k_scan<<<1, 128, 0, stream>>>(G, S);
    k_emit<<<1, NCHUNK, 0, stream>>>(b, S, h, out);
}